// Model_61040075210793
// MI455X (gfx1250) — compile-verified
//
#include <hip/hip_runtime.h>
#include <math.h>

typedef __attribute__((ext_vector_type(2))) float v2f;
typedef __attribute__((ext_vector_type(8))) float v8f;

// ---------------------------------------------------------------------------
// helpers
// ---------------------------------------------------------------------------
__device__ __forceinline__ float red16(float v) {      // sum within 16-lane halves
  v += __shfl_xor(v, 1, 32);
  v += __shfl_xor(v, 2, 32);
  v += __shfl_xor(v, 4, 32);
  v += __shfl_xor(v, 8, 32);
  return v;
}
__device__ __forceinline__ float red32(float v) {      // full-wave sum
  v = red16(v);
  v += __shfl_xor(v, 16, 32);
  return v;
}
__device__ __forceinline__ float redmax32(float v) {
  for (int k = 1; k < 32; k <<= 1) v = fmaxf(v, __shfl_xor(v, k, 32));
  return v;
}

// ---------------------------------------------------------------------------
// binning / histogram / scan / scatter
// ---------------------------------------------------------------------------
__global__ void k_hist(const int* __restrict__ key, int E, int* __restrict__ cnt) {
  int i = blockIdx.x * blockDim.x + threadIdx.x;
  if (i < E) atomicAdd(&cnt[key[i]], 1);
}

__global__ void k_scan_rel(const int* __restrict__ cnt, int* __restrict__ binStart,
                           int* __restrict__ tileOff) {
  if (blockIdx.x == 0 && threadIdx.x == 0) {
    int eo = 0, to = 0;
    for (int r = 0; r < 16; ++r) {
      binStart[r] = eo;
      tileOff[r]  = to;
      eo += cnt[r];
      to += (cnt[r] + 15) >> 4;
    }
    binStart[16] = eo;
    tileOff[16]  = to;
  }
}

__global__ void k_scatter(const int* __restrict__ key, int E,
                          const int* __restrict__ start, int* __restrict__ cur,
                          int* __restrict__ outIdx) {
  int i = blockIdx.x * blockDim.x + threadIdx.x;
  if (i < E) {
    int k = key[i];
    int p = atomicAdd(&cur[k], 1);
    outIdx[start[k] + p] = i;
  }
}

__global__ void k_scan_blk(const int* __restrict__ cnt, int n,
                           int* __restrict__ excl, int* __restrict__ blkSum) {
  __shared__ int s[256];
  int i = blockIdx.x * 256 + threadIdx.x;
  int v = (i < n) ? cnt[i] : 0;
  s[threadIdx.x] = v;
  __syncthreads();
  for (int off = 1; off < 256; off <<= 1) {
    int t = (threadIdx.x >= off) ? s[threadIdx.x - off] : 0;
    __syncthreads();
    s[threadIdx.x] += t;
    __syncthreads();
  }
  if (i < n) excl[i] = s[threadIdx.x] - v;
  if (threadIdx.x == 255) blkSum[blockIdx.x] = s[255];
}

__global__ void k_scan_top(int* __restrict__ blkSum, int nb) {
  __shared__ int s[1024];
  int v = (threadIdx.x < nb) ? blkSum[threadIdx.x] : 0;
  s[threadIdx.x] = v;
  __syncthreads();
  for (int off = 1; off < 1024; off <<= 1) {
    int t = (threadIdx.x >= off) ? s[threadIdx.x - off] : 0;
    __syncthreads();
    s[threadIdx.x] += t;
    __syncthreads();
  }
  if (threadIdx.x < nb) blkSum[threadIdx.x] = s[threadIdx.x] - v;  // exclusive
}

__global__ void k_scan_add(int* __restrict__ excl, int n,
                           const int* __restrict__ blkSum, int E) {
  int i = blockIdx.x * blockDim.x + threadIdx.x;
  if (i < n) excl[i] += blkSum[i >> 8];
  if (i == 0) excl[n] = E;
}

// ---------------------------------------------------------------------------
// attention: per 16-edge tile of one relation, T = Hsrc*W, H = Hdst*W via WMMA,
// att = sum_n T[m,n] * tanh(H[m,n] + rel[n])
// ---------------------------------------------------------------------------
__global__ void __launch_bounds__(256)
k_attention(const int* __restrict__ ebin, const int* __restrict__ binStart,
            const int* __restrict__ tileOff, const int* __restrict__ esrc,
            const int* __restrict__ edst, const float* __restrict__ emb,
            const float* __restrict__ rel, const float* __restrict__ WR,
            float* __restrict__ att) {
  int wid  = (blockIdx.x * blockDim.x + threadIdx.x) >> 5;
  int lane = threadIdx.x & 31;
  int nTiles = tileOff[16];
  if (wid >= nTiles) return;

  int r = 0;
  while (r < 15 && tileOff[r + 1] <= wid) ++r;
  int e0   = binStart[r] + (wid - tileOff[r]) * 16;
  int eEnd = binStart[r + 1];

  int m  = lane & 15;
  int kh = lane >> 4;                       // which 16-lane half
  int myE = e0 + m;
  int eIdx = ebin[(myE < eEnd) ? myE : e0];
  const float* srcRow = emb + (long long)esrc[eIdx] * 64;
  const float* dstRow = emb + (long long)edst[eIdx] * 64;

  // preload A fragments for all 16 K-steps (K = 64)
  v2f as[16], ad[16];
#pragma unroll
  for (int ks = 0; ks < 16; ++ks) {
    int kk = ks * 4 + kh * 2;
    as[ks] = *(const v2f*)(srcRow + kk);
    ad[ks] = *(const v2f*)(dstRow + kk);
  }

  const float* Wr   = WR + (long long)r * 64 * 64;
  const float* relr = rel + r * 64;

  v8f acc = {};
#pragma unroll
  for (int nt = 0; nt < 4; ++nt) {
    int ncol = nt * 16 + m;
    v8f ct = {}, ch = {};
#pragma unroll
    for (int ks = 0; ks < 16; ++ks) {
      int kk = ks * 4 + kh * 2;
      v2f b;
      b.x = Wr[kk * 64 + ncol];
      b.y = Wr[(kk + 1) * 64 + ncol];
      ct = __builtin_amdgcn_wmma_f32_16x16x4_f32(false, as[ks], false, b,
                                                 (short)0, ct, false, false);
      ch = __builtin_amdgcn_wmma_f32_16x16x4_f32(false, ad[ks], false, b,
                                                 (short)0, ch, false, false);
    }
    float re = relr[ncol];
#pragma unroll
    for (int v = 0; v < 8; ++v) acc[v] += ct[v] * tanhf(ch[v] + re);
  }

  // reduce across N (16 lanes per half); row M = v + 8*kh
#pragma unroll
  for (int v = 0; v < 8; ++v) {
    float s = red16(acc[v]);
    int row  = v + 8 * kh;
    int eRow = e0 + row;
    if ((lane & 15) == 0 && eRow < eEnd) att[ebin[eRow]] = s;
  }
}

// ---------------------------------------------------------------------------
// edge softmax per dst node (wave per node, CSR gather, no atomics)
// ---------------------------------------------------------------------------
__global__ void __launch_bounds__(256)
k_softmax(const int* __restrict__ csr_off, const int* __restrict__ csr_e,
          float* __restrict__ attw, int n) {
  int wid  = (blockIdx.x * blockDim.x + threadIdx.x) >> 5;
  int lane = threadIdx.x & 31;
  if (wid >= n) return;
  int b = csr_off[wid], e = csr_off[wid + 1];
  float mx = -__builtin_inff();
  for (int j = b + lane; j < e; j += 32) mx = fmaxf(mx, attw[csr_e[j]]);
  mx = redmax32(mx);
  float sum = 0.f;
  for (int j = b + lane; j < e; j += 32) sum += expf(attw[csr_e[j]] - mx);
  sum = red32(sum);
  float inv = (sum > 0.f) ? 1.f / sum : 0.f;
  for (int j = b + lane; j < e; j += 32) {
    int ei = csr_e[j];
    attw[ei] = expf(attw[ei] - mx) * inv;
  }
}

// ---------------------------------------------------------------------------
// neighbor aggregation: h_nb[n] = sum_{e in in(n)} w[e] * h[src[e]]
// ---------------------------------------------------------------------------
template <int DIN>
__global__ void __launch_bounds__(256)
k_aggregate(const int* __restrict__ csr_off, const int* __restrict__ csr_e,
            const int* __restrict__ esrc, const float* __restrict__ w,
            const float* __restrict__ hin, float* __restrict__ hnb, int n) {
  int wid  = (blockIdx.x * blockDim.x + threadIdx.x) >> 5;
  int lane = threadIdx.x & 31;
  if (wid >= n) return;
  int b = csr_off[wid], e = csr_off[wid + 1];
  float acc0 = 0.f, acc1 = 0.f;
  for (int j = b; j < e; ++j) {
    int ei = csr_e[j];
    float ww = w[ei];
    const float* row = hin + (long long)esrc[ei] * DIN;
    acc0 += ww * row[lane];
    if (DIN == 64) acc1 += ww * row[lane + 32];
  }
  hnb[(long long)wid * DIN + lane] = acc0;
  if (DIN == 64) hnb[(long long)wid * DIN + lane + 32] = acc1;
}

// ---------------------------------------------------------------------------
// layer GEMM: out = leaky_relu((h .* h_nb) @ fc_w)  via WMMA f32 16x16x4
// ---------------------------------------------------------------------------
template <int DIN, int DOUT>
__global__ void __launch_bounds__(256)
k_layer_gemm(const float* __restrict__ hin, const float* __restrict__ hnb,
             const float* __restrict__ fw, float* __restrict__ hout, int n) {
  int wid  = (blockIdx.x * blockDim.x + threadIdx.x) >> 5;
  int lane = threadIdx.x & 31;
  int mtiles = (n + 15) >> 4;
  int ntiles = DOUT / 16;
  if (wid >= mtiles * ntiles) return;
  int mt = wid / ntiles, nt = wid % ntiles;
  int m = lane & 15, kh = lane >> 4;
  int node = mt * 16 + m;
  if (node >= n) node = n - 1;
  long long row = (long long)node * DIN;

  v2f a[DIN / 4];
#pragma unroll
  for (int ks = 0; ks < DIN / 4; ++ks) {
    int kk = ks * 4 + kh * 2;
    a[ks].x = hin[row + kk] * hnb[row + kk];
    a[ks].y = hin[row + kk + 1] * hnb[row + kk + 1];
  }
  int ncol = nt * 16 + m;
  v8f c = {};
#pragma unroll
  for (int ks = 0; ks < DIN / 4; ++ks) {
    int kk = ks * 4 + kh * 2;
    v2f b;
    b.x = fw[kk * DOUT + ncol];
    b.y = fw[(kk + 1) * DOUT + ncol];
    c = __builtin_amdgcn_wmma_f32_16x16x4_f32(false, a[ks], false, b, (short)0,
                                              c, false, false);
  }
#pragma unroll
  for (int v = 0; v < 8; ++v) {
    float x = c[v];
    x = (x > 0.f) ? x : 0.01f * x;
    int rr = mt * 16 + v + 8 * kh;
    if (rr < n) hout[(long long)rr * DOUT + ncol] = x;
  }
}

// ---------------------------------------------------------------------------
// per-row L2 normalize into output slab (stride 176)
// ---------------------------------------------------------------------------
template <int DOUT>
__global__ void __launch_bounds__(256)
k_l2norm_out(const float* __restrict__ h, float* __restrict__ out, int colOff,
             int n) {
  int wid  = (blockIdx.x * blockDim.x + threadIdx.x) >> 5;
  int lane = threadIdx.x & 31;
  if (wid >= n) return;
  const float* rowp = h + (long long)wid * DOUT;
  float v0 = (lane < DOUT) ? rowp[lane] : 0.f;
  float v1 = (DOUT == 64) ? rowp[lane + 32] : 0.f;
  float ss = red32(v0 * v0 + v1 * v1);
  float inv = 1.f / fmaxf(sqrtf(ss), 1e-12f);
  float* orow = out + (long long)wid * 176 + colOff;
  if (lane < DOUT) orow[lane] = v0 * inv;
  if (DOUT == 64) orow[lane + 32] = v1 * inv;
}

__global__ void k_copy_h0(const float* __restrict__ emb, float* __restrict__ out,
                          int n) {
  long long i = (long long)blockIdx.x * blockDim.x + threadIdx.x;
  if (i < (long long)n * 64) {
    long long r = i >> 6;
    int d = (int)(i & 63);
    out[r * 176 + d] = emb[i];
  }
}

// ---------------------------------------------------------------------------
// launch
// ---------------------------------------------------------------------------
static inline char* ws_take(char*& p, size_t bytes) {
  char* r = p;
  p += (bytes + 255) & ~(size_t)255;
  return r;
}

extern "C" void kernel_launch(void* const* d_in, const int* in_sizes, int n_in,
                              void* d_out, int out_size, void* d_ws,
                              size_t ws_size, hipStream_t stream) {
  (void)n_in; (void)out_size; (void)ws_size;
  int n = in_sizes[0];           // 100000 nodes
  int E = in_sizes[1];           // 3.2M edges
  const int*   esrc = (const int*)d_in[1];
  const int*   edst = (const int*)d_in[2];
  const int*   etyp = (const int*)d_in[3];
  const float* emb  = (const float*)d_in[4];
  const float* rel  = (const float*)d_in[5];
  const float* WR   = (const float*)d_in[6];
  const float* fc0  = (const float*)d_in[7];
  const float* fc1  = (const float*)d_in[8];
  const float* fc2  = (const float*)d_in[9];
  float* out = (float*)d_out;

  char* p = (char*)d_ws;
  float* attw    = (float*)ws_take(p, (size_t)E * 4);
  int*   ibin    = (int*)  ws_take(p, (size_t)E * 4);
  int*   csr     = (int*)  ws_take(p, (size_t)E * 4);
  int*   csr_off = (int*)  ws_take(p, (size_t)(n + 1) * 4);
  int*   cntd    = (int*)  ws_take(p, (size_t)n * 4);     // counts then cursor
  int    nb      = (n + 255) / 256;
  int*   blkSum  = (int*)  ws_take(p, (size_t)nb * 4);
  int*   meta    = (int*)  ws_take(p, 128 * 4);
  int* cnt_rel  = meta + 0;   // 16
  int* cur_rel  = meta + 16;  // 16
  int* binStart = meta + 32;  // 17
  int* tileOff  = meta + 49;  // 17
  float* bufA = (float*)ws_take(p, (size_t)n * 64 * 4);
  float* bufB = (float*)ws_take(p, (size_t)n * 64 * 4);
  float* hnb  = (float*)ws_take(p, (size_t)n * 64 * 4);

  auto cdiv = [](long long a, long long b) { return (int)((a + b - 1) / b); };
  int eBlk = cdiv(E, 256);
  int nWav = cdiv((long long)n * 32, 256);

  hipMemsetAsync(meta, 0, 128 * sizeof(int), stream);
  hipMemsetAsync(cntd, 0, (size_t)n * sizeof(int), stream);

  // --- bin edges by relation, per-relation WMMA attention ---
  k_hist<<<eBlk, 256, 0, stream>>>(etyp, E, cnt_rel);
  k_scan_rel<<<1, 32, 0, stream>>>(cnt_rel, binStart, tileOff);
  k_scatter<<<eBlk, 256, 0, stream>>>(etyp, E, binStart, cur_rel, ibin);
  int maxTiles = E / 16 + 16;
  k_attention<<<cdiv((long long)maxTiles * 32, 256), 256, 0, stream>>>(
      ibin, binStart, tileOff, esrc, edst, emb, rel, WR, attw);

  // --- dst-CSR build ---
  k_hist<<<eBlk, 256, 0, stream>>>(edst, E, cntd);
  k_scan_blk<<<nb, 256, 0, stream>>>(cntd, n, csr_off, blkSum);
  k_scan_top<<<1, 1024, 0, stream>>>(blkSum, nb);
  k_scan_add<<<cdiv(n, 256), 256, 0, stream>>>(csr_off, n, blkSum, E);
  hipMemsetAsync(cntd, 0, (size_t)n * sizeof(int), stream);
  k_scatter<<<eBlk, 256, 0, stream>>>(edst, E, csr_off, cntd, csr);

  // --- edge softmax ---
  k_softmax<<<nWav, 256, 0, stream>>>(csr_off, csr, attw, n);

  // --- output col 0..63: raw h0 ---
  k_copy_h0<<<cdiv((long long)n * 64, 256), 256, 0, stream>>>(emb, out, n);

  int mt = (n + 15) / 16;
  // --- layer 0: 64 -> 64 ---
  k_aggregate<64><<<nWav, 256, 0, stream>>>(csr_off, csr, esrc, attw, emb, hnb, n);
  k_layer_gemm<64, 64><<<cdiv((long long)mt * 4 * 32, 256), 256, 0, stream>>>(
      emb, hnb, fc0, bufA, n);
  k_l2norm_out<64><<<nWav, 256, 0, stream>>>(bufA, out, 64, n);
  // --- layer 1: 64 -> 32 ---
  k_aggregate<64><<<nWav, 256, 0, stream>>>(csr_off, csr, esrc, attw, bufA, hnb, n);
  k_layer_gemm<64, 32><<<cdiv((long long)mt * 2 * 32, 256), 256, 0, stream>>>(
      bufA, hnb, fc1, bufB, n);
  k_l2norm_out<32><<<nWav, 256, 0, stream>>>(bufB, out, 128, n);
  // --- layer 2: 32 -> 16 ---
  k_aggregate<32><<<nWav, 256, 0, stream>>>(csr_off, csr, esrc, attw, bufB, hnb, n);
  k_layer_gemm<32, 16><<<cdiv((long long)mt * 1 * 32, 256), 256, 0, stream>>>(
      bufB, hnb, fc2, bufA, n);
  k_l2norm_out<16><<<nWav, 256, 0, stream>>>(bufA, out, 160, n);
}